// OOSSMBlock_59966333386870
// MI455X (gfx1250) — compile-verified
//
#include <hip/hip_runtime.h>
#include <hip/hip_bf16.h>

// ---- problem constants (match reference) ----
#define D_MODEL 512
#define D_INNER 1024
#define D_STATE 16
#define D_CONV  4
#define DT_RANK 32
#define B_SZ    2
#define L_SEQ   2048
#define ROWS    (B_SZ * L_SEQ)        // 4096
#define E_PROJ  (2 * D_INNER)         // 2048
#define DTBC_N  (DT_RANK + 2*D_STATE) // 64

typedef float v2f __attribute__((ext_vector_type(2)));
typedef float v8f __attribute__((ext_vector_type(8)));

// ---------------------------------------------------------------------------
// LayerNorm: one wave (32 lanes) per row of 512.
// ---------------------------------------------------------------------------
__global__ __launch_bounds__(256) void ln_kernel(const float* __restrict__ x,
                                                 const float* __restrict__ w,
                                                 const float* __restrict__ b,
                                                 float* __restrict__ xn) {
    int row  = (blockIdx.x * blockDim.x + threadIdx.x) >> 5;
    int lane = threadIdx.x & 31;
    if (row >= ROWS) return;
    const float* xr = x + row * D_MODEL;
    float s = 0.f, s2 = 0.f;
    for (int i = lane; i < D_MODEL; i += 32) { float v = xr[i]; s += v; s2 += v*v; }
    for (int m = 16; m; m >>= 1) { s += __shfl_xor(s, m); s2 += __shfl_xor(s2, m); }
    float mu = s * (1.f / D_MODEL);
    float var = s2 * (1.f / D_MODEL) - mu * mu;
    float rs = rsqrtf(var + 1e-5f);
    float* o = xn + row * D_MODEL;
    for (int i = lane; i < D_MODEL; i += 32) o[i] = (xr[i] - mu) * rs * w[i] + b[i];
}

// ---------------------------------------------------------------------------
// fp32 WMMA GEMM, LDS double-buffered:  O[m][n] = sum_k A[m][k] * W[n][k]
// Block: 8 waves -> 128(M) x 64(N); K chunk = 32 (8 WMMA K-steps of 4).
// LDS tiles stride 36 floats: 16B-aligned rows for b128 stores, and
// conflict-free ds_load_b64 fragment reads (36*lm mod 64 distinct, h adds +2).
//   EPI 0: plain   EPI 1: softplus(acc + bias[n])   EPI 2: acc + resid[m][n]
// ---------------------------------------------------------------------------
#define KC        32
#define LDSS      36
#define A_TILE    (128 * LDSS)   // 4608 floats
#define B_TILE    (64  * LDSS)   // 2304 floats

template <int EPI>
__global__ __launch_bounds__(256) void gemm_wmma_kernel(
    const float* __restrict__ A, const float* __restrict__ W,
    float* __restrict__ O, int K, int lda, int ldw, int ldo,
    const float* __restrict__ bias, const float* __restrict__ resid) {

    __shared__ float sA[2 * A_TILE];
    __shared__ float sB[2 * B_TILE];

    const int tid  = threadIdx.x;
    const int lane = tid & 31;
    const int wave = tid >> 5;
    const int half = lane >> 4;      // 0 or 1
    const int lm   = lane & 15;

    const int row0 = blockIdx.y * 128;
    const int col0 = blockIdx.x * 64;

    v8f acc0 = {}, acc1 = {}, acc2 = {}, acc3 = {};

    // Cooperative staging: 8 threads per row, each a float4; A=4 passes of 32
    // rows (128 rows), B=2 passes (64 rows).
    const int srow = tid >> 3;          // 0..31
    const int sq   = (tid & 7) * 4;     // 0,4,...,28
    const float* gA = A + (row0 + srow) * lda + sq;
    const float* gB = W + (col0 + srow) * ldw + sq;

    float4 ra[4], rb[2];
    #pragma unroll
    for (int p = 0; p < 4; ++p) ra[p] = *(const float4*)(gA + p * 32 * lda);
    #pragma unroll
    for (int p = 0; p < 2; ++p) rb[p] = *(const float4*)(gB + p * 32 * ldw);

    const int nchunk = K / KC;
    for (int c = 0; c < nchunk; ++c) {
        float* As = sA + (c & 1) * A_TILE;
        float* Bs = sB + (c & 1) * B_TILE;

        // 1) commit staged chunk c to LDS
        #pragma unroll
        for (int p = 0; p < 4; ++p)
            *(float4*)&As[(p * 32 + srow) * LDSS + sq] = ra[p];
        #pragma unroll
        for (int p = 0; p < 2; ++p)
            *(float4*)&Bs[(p * 32 + srow) * LDSS + sq] = rb[p];
        __syncthreads();

        // 2) issue global loads for chunk c+1 (overlap with compute below)
        if (c + 1 < nchunk) {
            const int k1 = (c + 1) * KC;
            #pragma unroll
            for (int p = 0; p < 4; ++p) ra[p] = *(const float4*)(gA + p * 32 * lda + k1);
            #pragma unroll
            for (int p = 0; p < 2; ++p) rb[p] = *(const float4*)(gB + p * 32 * ldw + k1);
        }

        // 3) compute chunk c from LDS: 8 K-steps x 4 WMMAs
        const float* ab = As + (wave * 16 + lm) * LDSS + 2 * half;
        const float* b0 = Bs + ( 0 + lm) * LDSS + 2 * half;
        const float* b1 = Bs + (16 + lm) * LDSS + 2 * half;
        const float* b2 = Bs + (32 + lm) * LDSS + 2 * half;
        const float* b3 = Bs + (48 + lm) * LDSS + 2 * half;
        #pragma unroll
        for (int ks = 0; ks < 8; ++ks) {
            const int kk = ks * 4;
            v2f a  = *(const v2f*)(ab + kk);
            v2f v0 = *(const v2f*)(b0 + kk);
            v2f v1 = *(const v2f*)(b1 + kk);
            v2f v2 = *(const v2f*)(b2 + kk);
            v2f v3 = *(const v2f*)(b3 + kk);
            acc0 = __builtin_amdgcn_wmma_f32_16x16x4_f32(false, a, false, v0, (short)0, acc0, false, false);
            acc1 = __builtin_amdgcn_wmma_f32_16x16x4_f32(false, a, false, v1, (short)0, acc1, false, false);
            acc2 = __builtin_amdgcn_wmma_f32_16x16x4_f32(false, a, false, v2, (short)0, acc2, false, false);
            acc3 = __builtin_amdgcn_wmma_f32_16x16x4_f32(false, a, false, v3, (short)0, acc3, false, false);
        }
        __syncthreads();  // all waves done reading buf before it is refilled
    }

    // Store: VGPR r -> row = row0 + wave*16 + r + 8*half, col = col0 + nt*16 + lm
    const int rbase = row0 + wave * 16;
    #pragma unroll
    for (int r = 0; r < 8; ++r) {
        int row = rbase + r + 8 * half;
        float* orow = O + row * ldo;
        float v[4] = { acc0[r], acc1[r], acc2[r], acc3[r] };
        #pragma unroll
        for (int nt = 0; nt < 4; ++nt) {
            int col = col0 + nt * 16 + lm;
            float val = v[nt];
            if (EPI == 1) {              // dt_proj bias + softplus
                val += bias[col];
                val = (val > 20.f) ? val : log1pf(expf(val));
            } else if (EPI == 2) {       // out_proj + residual
                val += resid[row * ldo + col];
            }
            orow[col] = val;
        }
    }
}

// ---------------------------------------------------------------------------
// Depthwise causal conv (width 4) + bias + SiLU. One thread per (b,l,c).
// ---------------------------------------------------------------------------
__global__ __launch_bounds__(256) void conv_silu_kernel(const float* __restrict__ xz,
                                                        const float* __restrict__ cw,
                                                        const float* __restrict__ cb,
                                                        float* __restrict__ xf) {
    int idx = blockIdx.x * blockDim.x + threadIdx.x;   // (b*L + l)*1024 + c
    int c = idx & (D_INNER - 1);
    int l = (idx >> 10) & (L_SEQ - 1);
    int b = idx >> 21;
    int rb = b * L_SEQ;
    float s = cb[c];
    const float* w = cw + c * D_CONV;
    #pragma unroll
    for (int t = 0; t < D_CONV; ++t) {
        int ll = l - (D_CONV - 1) + t;
        if (ll >= 0) s += w[t] * xz[(rb + ll) * E_PROJ + c];
    }
    float sig = 1.f / (1.f + expf(-s));
    xf[idx] = s * sig;
}

// ---------------------------------------------------------------------------
// Selective scan: one lane per (b, channel d, state s); 16-lane shfl reduce.
// y written in-place over x_flat (all lanes read x before lane s==0 writes).
// ---------------------------------------------------------------------------
__global__ __launch_bounds__(256) void scan_kernel(const float* __restrict__ dt,
                                                   const float* __restrict__ xf,
                                                   const float* __restrict__ dtBC,
                                                   const float* __restrict__ xz,
                                                   const float* __restrict__ A_log,
                                                   const float* __restrict__ Dvec,
                                                   float* __restrict__ y) {
    int tid = blockIdx.x * blockDim.x + threadIdx.x;   // 32768 = B * 1024 * 16
    int s   = tid & (D_STATE - 1);
    int dch = (tid >> 4) & (D_INNER - 1);
    int b   = tid >> 14;

    float Aval = -expf(A_log[dch * D_STATE + s]);
    float Dd   = Dvec[dch];
    float h    = 0.f;
    const int rowbase = b * L_SEQ;

    for (int l = 0; l < L_SEQ; ++l) {
        int row = rowbase + l;
        float dtv = dt[row * D_INNER + dch];
        float xv  = xf[row * D_INNER + dch];
        float Bv  = dtBC[row * DTBC_N + DT_RANK + s];
        float Cv  = dtBC[row * DTBC_N + DT_RANK + D_STATE + s];
        float dA  = expf(dtv * Aval);
        h = h * dA + dtv * Bv * xv;
        float yp = h * Cv;
        yp += __shfl_xor(yp, 8);
        yp += __shfl_xor(yp, 4);
        yp += __shfl_xor(yp, 2);
        yp += __shfl_xor(yp, 1);
        if (s == 0) {
            float zv  = xz[row * E_PROJ + D_INNER + dch];
            float sig = 1.f / (1.f + expf(-zv));
            y[row * D_INNER + dch] = (yp + xv * Dd) * (zv * sig);
        }
    }
}

// ---------------------------------------------------------------------------
extern "C" void kernel_launch(void* const* d_in, const int* in_sizes, int n_in,
                              void* d_out, int out_size, void* d_ws, size_t ws_size,
                              hipStream_t stream) {
    const float* x         = (const float*)d_in[0];
    const float* ln_w      = (const float*)d_in[1];
    const float* ln_b      = (const float*)d_in[2];
    const float* in_proj_w = (const float*)d_in[3];
    const float* conv_w    = (const float*)d_in[4];
    const float* conv_b    = (const float*)d_in[5];
    const float* x_proj_w  = (const float*)d_in[6];
    const float* dt_proj_w = (const float*)d_in[7];
    const float* dt_proj_b = (const float*)d_in[8];
    const float* A_log     = (const float*)d_in[9];
    const float* Dvec      = (const float*)d_in[10];
    const float* out_proj_w= (const float*)d_in[11];
    float* out = (float*)d_out;

    // Workspace layout (floats)
    float* xn   = (float*)d_ws;                    // 4096 x 512
    float* xz   = xn   + ROWS * D_MODEL;           // 4096 x 2048
    float* xf   = xz   + ROWS * E_PROJ;            // 4096 x 1024 (x_flat, later y)
    float* dtbc = xf   + ROWS * D_INNER;           // 4096 x 64
    float* dtb  = dtbc + ROWS * DTBC_N;            // 4096 x 1024

    // 1) LayerNorm (8 rows per 256-thread block)
    ln_kernel<<<ROWS / 8, 256, 0, stream>>>(x, ln_w, ln_b, xn);

    // 2) in_proj: [4096,512] x [2048,512]^T -> [4096,2048]
    gemm_wmma_kernel<0><<<dim3(E_PROJ / 64, ROWS / 128), 256, 0, stream>>>(
        xn, in_proj_w, xz, D_MODEL, D_MODEL, D_MODEL, E_PROJ, nullptr, nullptr);

    // 3) depthwise conv + SiLU -> x_flat
    conv_silu_kernel<<<(ROWS * D_INNER) / 256, 256, 0, stream>>>(xz, conv_w, conv_b, xf);

    // 4) x_proj: [4096,1024] x [64,1024]^T -> [4096,64]
    gemm_wmma_kernel<0><<<dim3(DTBC_N / 64, ROWS / 128), 256, 0, stream>>>(
        xf, x_proj_w, dtbc, D_INNER, D_INNER, D_INNER, DTBC_N, nullptr, nullptr);

    // 5) dt_proj + softplus: [4096,32(lda=64)] x [1024,32]^T -> [4096,1024]
    gemm_wmma_kernel<1><<<dim3(D_INNER / 64, ROWS / 128), 256, 0, stream>>>(
        dtbc, dt_proj_w, dtb, DT_RANK, DTBC_N, DT_RANK, D_INNER, dt_proj_b, nullptr);

    // 6) selective scan + D skip + SiLU(z) gate -> y (in-place over x_flat)
    scan_kernel<<<(B_SZ * D_INNER * D_STATE) / 256, 256, 0, stream>>>(
        dtb, xf, dtbc, xz, A_log, Dvec, xf);

    // 7) out_proj + residual: [4096,1024] x [512,1024]^T -> [4096,512]
    gemm_wmma_kernel<2><<<dim3(D_MODEL / 64, ROWS / 128), 256, 0, stream>>>(
        xf, out_proj_w, out, D_INNER, D_INNER, D_INNER, D_MODEL, nullptr, x);

    (void)in_sizes; (void)n_in; (void)out_size; (void)ws_size;
}